// GAT_2164663517732
// MI455X (gfx1250) — compile-verified
//
#include <hip/hip_runtime.h>
#include <hip/hip_bf16.h>
#include <math.h>

// ---------------------------------------------------------------------------
// Types for WMMA fragments (fp32 path: V_WMMA_F32_16X16X4_F32)
// ---------------------------------------------------------------------------
typedef float v2f __attribute__((ext_vector_type(2)));
typedef float v8f __attribute__((ext_vector_type(8)));

// ---------------------------------------------------------------------------
// Order-preserving float <-> uint key for atomicMax-based scatter-max.
// key(0u) corresponds to -NaN, so zero-initialized keys act as -infinity.
// ---------------------------------------------------------------------------
__device__ __forceinline__ unsigned fkey(float f) {
  unsigned b = __float_as_uint(f);
  return (b & 0x80000000u) ? ~b : (b | 0x80000000u);
}
__device__ __forceinline__ float fdec(unsigned k) {
  unsigned b = (k & 0x80000000u) ? (k ^ 0x80000000u) : ~k;
  return __uint_as_float(b);
}

__device__ __forceinline__ float lrelu(float a) {
  return (a > 0.0f) ? a : 0.2f * a;
}

// ---------------------------------------------------------------------------
// Fill kernels (d_out / workspace are poisoned by the harness)
// ---------------------------------------------------------------------------
__global__ void fill_f32(float* __restrict__ p, float v, size_t n) {
  size_t i = (size_t)blockIdx.x * blockDim.x + threadIdx.x;
  if (i < n) p[i] = v;
}
__global__ void fill_u32(unsigned* __restrict__ p, unsigned v, size_t n) {
  size_t i = (size_t)blockIdx.x * blockDim.x + threadIdx.x;
  if (i < n) p[i] = v;
}

// ---------------------------------------------------------------------------
// Row-major fp32 GEMM with V_WMMA_F32_16X16X4_F32.
// C[M,Nc] = A[M,K] * B[K,Nc].  Requires M%16==0, K%4==0, Nc%64==0.
// One wave computes a 16x64 strip (4 accumulator tiles); A fragment reused 4x.
// All branches are wave-uniform so EXEC is all-ones at every WMMA.
// ---------------------------------------------------------------------------
__global__ void gemm_f32_wmma(const float* __restrict__ A,
                              const float* __restrict__ B,
                              float* __restrict__ C,
                              int M, int K, int Nc) {
  const int lane = threadIdx.x & 31;
  const int wave = threadIdx.y;
  const int colGroups = Nc >> 6;                     // 64-wide column groups
  const int numTiles  = (M >> 4) * colGroups;
  const int tile = blockIdx.x * blockDim.y + wave;   // wave-uniform
  if (tile >= numTiles) return;

  const int rowTile  = tile / colGroups;
  const int colGroup = tile % colGroups;

  const int half = lane >> 4;       // 0: K pair {k0,k0+1}, 1: {k0+2,k0+3}
  const int l    = lane & 15;
  const int arow = rowTile * 16 + l;

  const float* __restrict__ Arow = A + (size_t)arow * K;

  v8f acc[4] = {};

  for (int k0 = 0; k0 < K; k0 += 4) {
    const int ka = k0 + 2 * half;
    v2f a;
    a.x = Arow[ka];
    a.y = Arow[ka + 1];
#pragma unroll
    for (int cc = 0; cc < 4; ++cc) {
      const int bcol = (colGroup << 6) + (cc << 4) + l;
      v2f b;
      b.x = B[(size_t)ka * Nc + bcol];
      b.y = B[(size_t)(ka + 1) * Nc + bcol];
      acc[cc] = __builtin_amdgcn_wmma_f32_16x16x4_f32(
          /*neg_a=*/false, a, /*neg_b=*/false, b,
          /*c_mod=*/(short)0, acc[cc],
          /*reuse_a=*/false, /*reuse_b=*/false);
    }
  }

  // C/D layout: VGPR r -> row = 16*rowTile + r + 8*half, col = 64*colGroup + 16*cc + (lane&15)
  const int orow0 = rowTile * 16 + 8 * half;
#pragma unroll
  for (int cc = 0; cc < 4; ++cc) {
    const int ocol = (colGroup << 6) + (cc << 4) + l;
#pragma unroll
    for (int r = 0; r < 8; ++r) {
      C[(size_t)(orow0 + r) * Nc + ocol] = acc[cc][r];
    }
  }
}

// ---------------------------------------------------------------------------
// Per-node attention coefficients: a_s[n,h] = <H[n,h,:], att_src[h,:]>
// ---------------------------------------------------------------------------
__global__ void att_coef(const float* __restrict__ H,
                         const float* __restrict__ att_src,
                         const float* __restrict__ att_dst,
                         float* __restrict__ a_s, float* __restrict__ a_d,
                         int n_nodes, int heads, int C) {
  int idx = blockIdx.x * blockDim.x + threadIdx.x;   // node*heads + h
  if (idx >= n_nodes * heads) return;
  int n = idx / heads, h = idx % heads;
  const float* __restrict__ hp = H + (size_t)n * heads * C + (size_t)h * C;
  const float* __restrict__ as = att_src + (size_t)h * C;
  const float* __restrict__ ad = att_dst + (size_t)h * C;
  float s = 0.0f, d = 0.0f;
  for (int c = 0; c < C; ++c) {
    float v = hp[c];
    s += v * as[c];
    d += v * ad[c];
  }
  a_s[idx] = s;
  a_d[idx] = d;
}

// ---------------------------------------------------------------------------
// Edge pass 1: alpha = leaky_relu(a_s[src]+a_d[dst]); scatter-max into key[dst].
// Edges e < E come from edge_index; e >= E are self-loops.
// ---------------------------------------------------------------------------
__device__ __forceinline__ void edge_endpoints(const int* __restrict__ ei,
                                               int E, int e, int& s, int& d) {
  if (e < E) { s = ei[e]; d = ei[E + e]; } else { s = e - E; d = e - E; }
}

__global__ void edge_max(const int* __restrict__ ei, int E, int n_nodes,
                         int heads,
                         const float* __restrict__ a_s,
                         const float* __restrict__ a_d,
                         unsigned* __restrict__ mkey) {
  int e = blockIdx.x * blockDim.x + threadIdx.x;
  if (e >= E + n_nodes) return;
  int s, d;
  edge_endpoints(ei, E, e, s, d);
  for (int h = 0; h < heads; ++h) {
    float al = lrelu(a_s[s * heads + h] + a_d[d * heads + h]);
    atomicMax(&mkey[d * heads + h], fkey(al));
  }
}

// ---------------------------------------------------------------------------
// Edge pass 2: ea = exp(alpha - max[dst]); scatter-add into denom[dst].
// (alpha recomputed: 2 gathers + exp is cheaper than a per-edge buffer.)
// ---------------------------------------------------------------------------
__global__ void edge_expsum(const int* __restrict__ ei, int E, int n_nodes,
                            int heads,
                            const float* __restrict__ a_s,
                            const float* __restrict__ a_d,
                            const unsigned* __restrict__ mkey,
                            float* __restrict__ denom) {
  int e = blockIdx.x * blockDim.x + threadIdx.x;
  if (e >= E + n_nodes) return;
  int s, d;
  edge_endpoints(ei, E, e, s, d);
  for (int h = 0; h < heads; ++h) {
    float al = lrelu(a_s[s * heads + h] + a_d[d * heads + h]);
    float ea = expf(al - fdec(mkey[d * heads + h]));
    atomicAdd(&denom[d * heads + h], ea);
  }
}

// ---------------------------------------------------------------------------
// Edge pass 3: out[dst] += (ea/denom[dst]) * H[src].
// One block per edge, one thread per (head,channel). Gather + atomic-add are
// L2-resident (feature matrices fit in the 192MB L2).
// ---------------------------------------------------------------------------
__global__ void edge_aggregate(const int* __restrict__ ei, int E, int n_nodes,
                               int heads, int C,
                               const float* __restrict__ a_s,
                               const float* __restrict__ a_d,
                               const unsigned* __restrict__ mkey,
                               const float* __restrict__ denom,
                               const float* __restrict__ H,
                               float* __restrict__ out) {
  int e = blockIdx.x;
  if (e >= E + n_nodes) return;
  int s, d;
  edge_endpoints(ei, E, e, s, d);
  int t = threadIdx.x;              // t = h*C + c ; blockDim.x == heads*C
  int h = t / C;
  float al = lrelu(a_s[s * heads + h] + a_d[d * heads + h]);   // wave-uniform scalars
  float ea = expf(al - fdec(mkey[d * heads + h]));
  float w  = ea / (denom[d * heads + h] + 1e-16f);
  atomicAdd(&out[(size_t)d * heads * C + t], w * H[(size_t)s * heads * C + t]);
}

// ---------------------------------------------------------------------------
// out[i] = act(out[i] + bias[i % cols]);  act = ELU (layer 1) or identity.
// ---------------------------------------------------------------------------
__global__ void bias_act(float* __restrict__ X, const float* __restrict__ b,
                         size_t total, int cols, int do_elu) {
  size_t i = (size_t)blockIdx.x * blockDim.x + threadIdx.x;
  if (i >= total) return;
  float v = X[i] + b[i % cols];
  if (do_elu) v = (v > 0.0f) ? v : (expf(v) - 1.0f);
  X[i] = v;
}

// ---------------------------------------------------------------------------
static inline unsigned cdiv(size_t a, size_t b) {
  return (unsigned)((a + b - 1) / b);
}

extern "C" void kernel_launch(void* const* d_in, const int* in_sizes, int n_in,
                              void* d_out, int out_size, void* d_ws, size_t ws_size,
                              hipStream_t stream) {
  const float* x   = (const float*)d_in[0];
  const int*   ei  = (const int*)  d_in[1];
  const float* W1  = (const float*)d_in[2];
  const float* as1 = (const float*)d_in[3];
  const float* ad1 = (const float*)d_in[4];
  const float* b1  = (const float*)d_in[5];
  const float* W2  = (const float*)d_in[6];
  const float* as2 = (const float*)d_in[7];
  const float* ad2 = (const float*)d_in[8];
  const float* b2  = (const float*)d_in[9];

  const int IN_CH = 128, HID = 64, HEADS = 4, OUT_CH = 64;
  const int N  = in_sizes[0] / IN_CH;
  const int E  = in_sizes[1] / 2;
  const int Et = E + N;                 // edges incl. self-loops
  const int F1 = HEADS * HID;           // 256

  // ---- workspace carve-up (fp32 words) ----
  float* ws = (float*)d_ws;
  float*    H1   = ws;            ws += (size_t)N * F1;      // x @ W1
  float*    AGG1 = ws;            ws += (size_t)N * F1;      // layer-1 aggregation
  float*    s1   = ws;            ws += (size_t)N * HEADS;
  float*    d1   = ws;            ws += (size_t)N * HEADS;
  unsigned* mk1  = (unsigned*)ws; ws += (size_t)N * HEADS;
  float*    den1 = ws;            ws += (size_t)N * HEADS;
  float*    H2   = ws;            ws += (size_t)N * OUT_CH;  // h @ W2
  float*    s2   = ws;            ws += (size_t)N;
  float*    d2v  = ws;            ws += (size_t)N;
  unsigned* mk2  = (unsigned*)ws; ws += (size_t)N;
  float*    den2 = ws;            ws += (size_t)N;
  float*    out  = (float*)d_out;

  // ---- zero-init accumulation targets ----
  fill_f32<<<cdiv((size_t)N * F1, 256), 256, 0, stream>>>(AGG1, 0.0f, (size_t)N * F1);
  fill_u32<<<cdiv((size_t)N * HEADS, 256), 256, 0, stream>>>(mk1, 0u, (size_t)N * HEADS);
  fill_f32<<<cdiv((size_t)N * HEADS, 256), 256, 0, stream>>>(den1, 0.0f, (size_t)N * HEADS);
  fill_u32<<<cdiv((size_t)N, 256), 256, 0, stream>>>(mk2, 0u, (size_t)N);
  fill_f32<<<cdiv((size_t)N, 256), 256, 0, stream>>>(den2, 0.0f, (size_t)N);
  fill_f32<<<cdiv((size_t)N * OUT_CH, 256), 256, 0, stream>>>(out, 0.0f, (size_t)N * OUT_CH);

  // ---- layer 1 ----
  {
    int tiles = (N / 16) * (F1 / 64);   // 16x64 strips per wave
    gemm_f32_wmma<<<cdiv(tiles, 8), dim3(32, 8), 0, stream>>>(x, W1, H1, N, IN_CH, F1);
  }
  att_coef<<<cdiv((size_t)N * HEADS, 256), 256, 0, stream>>>(H1, as1, ad1, s1, d1, N, HEADS, HID);
  edge_max<<<cdiv(Et, 256), 256, 0, stream>>>(ei, E, N, HEADS, s1, d1, mk1);
  edge_expsum<<<cdiv(Et, 256), 256, 0, stream>>>(ei, E, N, HEADS, s1, d1, mk1, den1);
  edge_aggregate<<<Et, HEADS * HID, 0, stream>>>(ei, E, N, HEADS, HID, s1, d1, mk1, den1, H1, AGG1);
  bias_act<<<cdiv((size_t)N * F1, 256), 256, 0, stream>>>(AGG1, b1, (size_t)N * F1, F1, /*elu=*/1);

  // ---- layer 2 ----
  {
    int tiles = (N / 16) * (OUT_CH / 64);
    gemm_f32_wmma<<<cdiv(tiles, 8), dim3(32, 8), 0, stream>>>(AGG1, W2, H2, N, F1, OUT_CH);
  }
  att_coef<<<cdiv((size_t)N, 256), 256, 0, stream>>>(H2, as2, ad2, s2, d2v, N, 1, OUT_CH);
  edge_max<<<cdiv(Et, 256), 256, 0, stream>>>(ei, E, N, 1, s2, d2v, mk2);
  edge_expsum<<<cdiv(Et, 256), 256, 0, stream>>>(ei, E, N, 1, s2, d2v, mk2, den2);
  edge_aggregate<<<Et, OUT_CH, 0, stream>>>(ei, E, N, 1, OUT_CH, s2, d2v, mk2, den2, H2, out);
  bias_act<<<cdiv((size_t)N * OUT_CH, 256), 256, 0, stream>>>(out, b2, (size_t)N * OUT_CH, OUT_CH, /*elu=*/0);
}